// Net_27711128994265
// MI455X (gfx1250) — compile-verified
//
#include <hip/hip_runtime.h>
#include <hip/hip_bf16.h>

#define N_NODES 10000
#define N_EDGES 640000
#define IN_CH   128
#define HID     128
#define OUT_CH  2

typedef __attribute__((ext_vector_type(2))) float v2f;
typedef __attribute__((ext_vector_type(8))) float v8f;

// ---------------------------------------------------------------------------
// Kernel 1: zero the workspace (agg, deg, hmax)
// ---------------------------------------------------------------------------
__global__ void k_init(float* __restrict__ ws, int total) {
    int i = blockIdx.x * blockDim.x + threadIdx.x;
    int stride = gridDim.x * blockDim.x;
    for (; i < total; i += stride) ws[i] = 0.0f;
}

// ---------------------------------------------------------------------------
// Kernel 2: edge scatter-add.  One wave per edge iteration:
//   lane l loads float4 #l of x[src] (512B coalesced, L2-resident) and does
//   4x global_atomic_add_f32 into agg[dst]; lane 0 counts degree.
// ---------------------------------------------------------------------------
__global__ void k_edge_scatter(const float* __restrict__ x,
                               const long long* __restrict__ ei,
                               float* __restrict__ agg,
                               float* __restrict__ deg) {
    const int lane = threadIdx.x & 31;
    const int gw   = (blockIdx.x * blockDim.x + threadIdx.x) >> 5;
    const int nw   = (gridDim.x * blockDim.x) >> 5;
    for (int e = gw; e < N_EDGES; e += nw) {
        const int src = (int)ei[e];
        const int dst = (int)ei[N_EDGES + e];
        const float4 v = ((const float4*)(x + (size_t)src * IN_CH))[lane];
        float* a = agg + (size_t)dst * IN_CH + lane * 4;
        atomicAdd(a + 0, v.x);
        atomicAdd(a + 1, v.y);
        atomicAdd(a + 2, v.z);
        atomicAdd(a + 3, v.w);
        if (lane == 0) atomicAdd(deg + dst, 1.0f);
    }
}

// ---------------------------------------------------------------------------
// Kernel 3: h = relu((agg/deg) @ W_l + b_l + x @ W_r), column-max over nodes.
// Block = 256 threads = 8 waves; block owns a 16-row node tile, wave w owns
// output columns [16w, 16w+16).  Each wave accumulates its 16x16 tile with
// v_wmma_f32_16x16x4_f32 (K=4 per op), 32 ops per 128-deep GEMM, two GEMMs.
//
// A (16x4 f32) layout: lanes 0-15 -> M=0..15 with K={k,k+1}; lanes 16-31 ->
//   same M with K={k+2,k+3}.   B (4x16) mirrors this with N across lanes.
// C/D: VGPR r -> row (r + 8*(lane>=16)), col = n0 + (lane&15).
// ---------------------------------------------------------------------------
__global__ void k_sage_wmma(const float* __restrict__ x,
                            const float* __restrict__ agg,
                            const float* __restrict__ deg,
                            const float* __restrict__ Wl,
                            const float* __restrict__ bl,
                            const float* __restrict__ Wr,
                            float* __restrict__ hmax) {
    const int wave = threadIdx.x >> 5;
    const int lane = threadIdx.x & 31;
    const int m16  = lane & 15;
    const int off  = (lane >> 4) << 1;      // 0 for lanes 0-15, 2 for 16-31
    const int n0   = wave * 16;
    const int rowA = blockIdx.x * 16 + m16; // node index this lane loads A for

    float s = deg[rowA];
    s = 1.0f / fmaxf(s, 1.0f);

    const float* fa = agg + (size_t)rowA * IN_CH;
    const float* fx = x   + (size_t)rowA * IN_CH;

    v8f c = {};

    // GEMM 1: (agg * inv_deg) @ W_l
    for (int k = 0; k < IN_CH; k += 4) {
        v2f a, b;
        a.x = fa[k + off]     * s;
        a.y = fa[k + off + 1] * s;
        b.x = Wl[(k + off)     * HID + n0 + m16];
        b.y = Wl[(k + off + 1) * HID + n0 + m16];
        c = __builtin_amdgcn_wmma_f32_16x16x4_f32(false, a, false, b,
                                                  (short)0, c, false, false);
    }
    // GEMM 2: x @ W_r  (accumulated into same C)
    for (int k = 0; k < IN_CH; k += 4) {
        v2f a, b;
        a.x = fx[k + off];
        a.y = fx[k + off + 1];
        b.x = Wr[(k + off)     * HID + n0 + m16];
        b.y = Wr[(k + off + 1) * HID + n0 + m16];
        c = __builtin_amdgcn_wmma_f32_16x16x4_f32(false, a, false, b,
                                                  (short)0, c, false, false);
    }

    // bias + relu + per-column max over this tile's 16 rows
    const float bias = bl[n0 + m16];
    float m = 0.0f;
    for (int r = 0; r < 8; ++r) {
        float v = fmaxf(c[r] + bias, 0.0f);
        m = fmaxf(m, v);
    }
    // lanes l and l+16 hold rows 0-7 / 8-15 of the same column -> combine
    m = fmaxf(m, __shfl_xor(m, 16, 32));
    if (lane < 16) {
        // relu output >= 0, so int-bit atomicMax == float max (hmax init 0)
        atomicMax((int*)hmax + n0 + m16, __float_as_int(m));
    }
}

// ---------------------------------------------------------------------------
// Kernel 4: MLP head.  news = relu(x[0]@W0+b0); h = relu([news,hmax]@W1+b1);
// out = h@W2+b2.  Single block of 128 threads; negligible cost.
// ---------------------------------------------------------------------------
__global__ void k_head(const float* __restrict__ x,
                       const float* __restrict__ hmax,
                       const float* __restrict__ W0, const float* __restrict__ b0,
                       const float* __restrict__ W1, const float* __restrict__ b1,
                       const float* __restrict__ W2, const float* __restrict__ b2,
                       float* __restrict__ out) {
    __shared__ float cat[2 * HID];
    __shared__ float h1[HID];
    const int t = threadIdx.x;   // 0..127

    float acc = b0[t];
    for (int k = 0; k < IN_CH; ++k) acc += x[k] * W0[k * HID + t];
    cat[t]       = fmaxf(acc, 0.0f);
    cat[HID + t] = hmax[t];      // written as int bits of a non-negative float
    __syncthreads();

    float a1 = b1[t];
    for (int k = 0; k < 2 * HID; ++k) a1 += cat[k] * W1[k * HID + t];
    h1[t] = fmaxf(a1, 0.0f);
    __syncthreads();

    if (t < OUT_CH) {
        float o = b2[t];
        for (int k = 0; k < HID; ++k) o += h1[k] * W2[k * OUT_CH + t];
        out[t] = o;
    }
}

// ---------------------------------------------------------------------------
extern "C" void kernel_launch(void* const* d_in, const int* in_sizes, int n_in,
                              void* d_out, int out_size, void* d_ws, size_t ws_size,
                              hipStream_t stream) {
    const float*     x  = (const float*)d_in[0];
    const long long* ei = (const long long*)d_in[1];
    const float*     Wl = (const float*)d_in[2];
    const float*     bl = (const float*)d_in[3];
    const float*     Wr = (const float*)d_in[4];
    const float*     W0 = (const float*)d_in[5];
    const float*     b0 = (const float*)d_in[6];
    const float*     W1 = (const float*)d_in[7];
    const float*     b1 = (const float*)d_in[8];
    const float*     W2 = (const float*)d_in[9];
    const float*     b2 = (const float*)d_in[10];
    float* out = (float*)d_out;

    // workspace layout: agg [N*128] | deg [N] | hmax [128]
    float* agg  = (float*)d_ws;
    float* deg  = agg + (size_t)N_NODES * IN_CH;
    float* hmax = deg + N_NODES;
    const int ws_floats = N_NODES * IN_CH + N_NODES + HID;

    k_init<<<(ws_floats + 255) / 256, 256, 0, stream>>>((float*)d_ws, ws_floats);

    k_edge_scatter<<<1280, 256, 0, stream>>>(x, ei, agg, deg);

    k_sage_wmma<<<N_NODES / 16, 256, 0, stream>>>(x, agg, deg, Wl, bl, Wr, hmax);

    k_head<<<1, HID, 0, stream>>>(x, hmax, W0, b0, W1, b1, W2, b2, out);
}